// Net_69518340653593
// MI455X (gfx1250) — compile-verified
//
#include <hip/hip_runtime.h>
#include <math.h>

// ---- problem constants (match reference) ----
#define N_NODES 50000
#define N_EDGES 1600000
#define D_IN    128
#define D_HID   50
#define D_OUT   10
#define M_TILES (N_NODES / 16)   // 3125 exact
#define WAVES_PER_BLK 8
#define NPAD 64                  // D_HID padded to 4 N-fragments of 16
#define KSTR 132                 // LDS k-stride (pad: (4n+k)%64 distinct per lane -> no bank conflicts)

typedef __attribute__((ext_vector_type(2))) float v2f;
typedef __attribute__((ext_vector_type(8))) float v8f;

// ---------------- degree: deg[row[e]] += 1 ----------------
__global__ void deg_kernel(const long long* __restrict__ row, float* __restrict__ deg) {
    int e = blockIdx.x * blockDim.x + threadIdx.x;
    if (e < N_EDGES) atomicAdd(&deg[(int)row[e]], 1.0f);
}

// ---------------- dinv = deg>0 ? rsqrt(max(deg,1)) : 0 ----------------
__global__ void dinv_kernel(const float* __restrict__ deg, float* __restrict__ dinv) {
    int n = blockIdx.x * blockDim.x + threadIdx.x;
    if (n < N_NODES) {
        float d = deg[n];
        dinv[n] = (d > 0.f) ? rsqrtf(fmaxf(d, 1.f)) : 0.f;
    }
}

// ---------------- w[e] = -dinv[row]*dinv[col] ----------------
__global__ void edgew_kernel(const long long* __restrict__ row,
                             const long long* __restrict__ col,
                             const float* __restrict__ dinv,
                             float* __restrict__ wE) {
    int e = blockIdx.x * blockDim.x + threadIdx.x;
    if (e < N_EDGES) wE[e] = -dinv[(int)row[e]] * dinv[(int)col[e]];
}

// ---------------- dst = -src (vectorized, N_NODES*D_IN floats) ----------------
__global__ void negcopy_kernel(const float4* __restrict__ src, float4* __restrict__ dst) {
    int i = blockIdx.x * blockDim.x + threadIdx.x;
    if (i < N_NODES * D_IN / 4) {
        float4 v = src[i];
        dst[i] = make_float4(-v.x, -v.y, -v.z, -v.w);
    }
}

// ---------------- scatter: dst[row] += scale*w[e]*h[col], one wave per edge ----------------
// lane handles 4 contiguous columns -> global_load_b128 gather + 4 fp32 L2 atomics
__global__ void scatter_kernel(const long long* __restrict__ row,
                               const long long* __restrict__ col,
                               const float* __restrict__ wE,
                               const float* __restrict__ h,
                               float* __restrict__ dst,
                               float scale) {
    int tid  = blockIdx.x * blockDim.x + threadIdx.x; // N_EDGES*32 = 51.2M < 2^31
    int e    = tid >> 5;
    int lane = tid & 31;
    long long r = row[e];
    long long c = col[e];
    float wgt = scale * wE[e];
    const float4 hv = *reinterpret_cast<const float4*>(h + c * D_IN + lane * 4);
    float* d = dst + r * D_IN + lane * 4;
    atomicAdd(d + 0, wgt * hv.x);
    atomicAdd(d + 1, wgt * hv.y);
    atomicAdd(d + 2, wgt * hv.z);
    atomicAdd(d + 3, wgt * hv.w);
}

// ---------------- C(N x 50) (+)= A(N x 128) @ B(128 x 50), fp32 WMMA ----------------
// B staged in LDS as sB[n*KSTR + k] (zero-padded for n>=50), so the inner loop is
// branch-free: ds_load_b64 for each B K-pair, global_load_b64 for each A K-pair,
// 4 v_wmma_f32_16x16x4_f32 per K-step. 8 waves/block, each owns a 16-row M-tile.
// Fragment layouts per ISA 7.12.2 (32-bit):
//   A 16x4 : lanes 0-15 -> M=lane, VGPR{0,1}=K{k0,k0+1}; lanes 16-31 -> K{k0+2,k0+3}
//   B 4x16 : lanes 0-15 -> N=lane, same half-wave K-pair split
//   C/D    : VGPR v -> row m0+v (+8 for upper half-wave), col n0+(lane&15)
__global__ __launch_bounds__(WAVES_PER_BLK * 32) void gemm_cheb(
        const float* __restrict__ A,
        const float* __restrict__ B,
        float* __restrict__ C,
        int beta) {
    __shared__ float sB[NPAD * KSTR];

    // cooperative staging: 64 x 128 transposed+padded copy of B
    for (int i = threadIdx.x; i < NPAD * D_IN; i += WAVES_PER_BLK * 32) {
        int n = i >> 7;       // 0..63
        int k = i & 127;
        sB[n * KSTR + k] = (n < D_HID) ? B[k * D_HID + n] : 0.f;
    }
    __syncthreads();

    const int lane  = threadIdx.x & 31;
    const int wave  = threadIdx.x >> 5;
    const int mtile = blockIdx.x * WAVES_PER_BLK + wave;
    if (mtile >= M_TILES) return;          // wave-uniform tail exit (no further syncs)
    const int m0   = mtile * 16;
    const int half = lane >> 4;            // selects K pair within fragment
    const int l    = lane & 15;

    // branch-free accumulator init: clamp index, mask value
    v8f acc[4];
    for (int t = 0; t < 4; ++t) {
        int n = t * 16 + l;
        int nc = (n < D_HID) ? n : 0;
        float mask = (beta != 0 && n < D_HID) ? 1.f : 0.f;
        for (int v = 0; v < 8; ++v)
            acc[t][v] = mask * C[(size_t)(m0 + half * 8 + v) * D_HID + nc];
    }

    const float* arow = A + (size_t)(m0 + l) * D_IN + half * 2;
#pragma unroll 4
    for (int k0 = 0; k0 < D_IN; k0 += 4) {
        v2f a;
        a[0] = arow[k0];
        a[1] = arow[k0 + 1];
        const float* bp = &sB[l * KSTR + k0 + half * 2];
#pragma unroll
        for (int t = 0; t < 4; ++t) {
            v2f b;
            b[0] = bp[t * 16 * KSTR + 0];
            b[1] = bp[t * 16 * KSTR + 1];
            acc[t] = __builtin_amdgcn_wmma_f32_16x16x4_f32(
                false, a, false, b, (short)0, acc[t], false, false);
        }
    }

    for (int t = 0; t < 4; ++t) {
        int n = t * 16 + l;
        if (n < D_HID) {
            for (int v = 0; v < 8; ++v)
                C[(size_t)(m0 + half * 8 + v) * D_HID + n] = acc[t][v];
        }
    }
}

// ---------------- head: relu(hid + cheb_b) @ fc_w + fc_b -> log_softmax ----------------
__global__ void head_kernel(const float* __restrict__ hid,
                            const float* __restrict__ cheb_b,
                            const float* __restrict__ fc_w,
                            const float* __restrict__ fc_b,
                            float* __restrict__ out) {
    __shared__ float sW[D_HID * D_OUT];
    __shared__ float scb[D_HID];
    __shared__ float sfb[D_OUT];
    for (int i = threadIdx.x; i < D_HID * D_OUT; i += blockDim.x) sW[i] = fc_w[i];
    for (int i = threadIdx.x; i < D_HID; i += blockDim.x) scb[i] = cheb_b[i];
    for (int i = threadIdx.x; i < D_OUT; i += blockDim.x) sfb[i] = fc_b[i];
    __syncthreads();

    int n = blockIdx.x * blockDim.x + threadIdx.x;
    if (n >= N_NODES) return;

    float z[D_OUT];
    for (int t = 0; t < D_OUT; ++t) z[t] = sfb[t];
    const float* hrow = hid + (size_t)n * D_HID;
    for (int j = 0; j < D_HID; ++j) {
        float h = fmaxf(hrow[j] + scb[j], 0.f);
        for (int t = 0; t < D_OUT; ++t) z[t] += h * sW[j * D_OUT + t];
    }
    float m = z[0];
    for (int t = 1; t < D_OUT; ++t) m = fmaxf(m, z[t]);
    float s = 0.f;
    for (int t = 0; t < D_OUT; ++t) s += __expf(z[t] - m);
    float lse = m + __logf(s);
    for (int t = 0; t < D_OUT; ++t) out[(size_t)n * D_OUT + t] = z[t] - lse;
}

extern "C" void kernel_launch(void* const* d_in, const int* in_sizes, int n_in,
                              void* d_out, int out_size, void* d_ws, size_t ws_size,
                              hipStream_t stream) {
    const float*     x      = (const float*)d_in[0];
    const long long* ei     = (const long long*)d_in[1];   // (2, E) int64
    const long long* row    = ei;
    const long long* col    = ei + N_EDGES;
    const float*     cheb_w = (const float*)d_in[2];       // (5, 128, 50)
    const float*     cheb_b = (const float*)d_in[3];
    const float*     fc_w   = (const float*)d_in[4];       // (50, 10)
    const float*     fc_b   = (const float*)d_in[5];
    float*           out    = (float*)d_out;

    // workspace layout (floats):
    float* deg  = (float*)d_ws;
    float* dinv = deg  + N_NODES;
    float* wE   = dinv + N_NODES;
    float* hid  = wE   + N_EDGES;                          // N x 50
    float* bufA = hid  + (size_t)N_NODES * D_HID;          // N x 128
    float* bufB = bufA + (size_t)N_NODES * D_IN;
    float* bufC = bufB + (size_t)N_NODES * D_IN;

    const int TPB = 256;
    const int EB  = (N_EDGES + TPB - 1) / TPB;             // 6250
    const int NB  = (N_NODES + TPB - 1) / TPB;             // 196
    const int SB  = (N_EDGES * 32) / TPB;                  // 200000 exact
    const int CB  = (N_NODES * D_IN / 4) / TPB;            // 6250 exact
    const int GB  = (M_TILES + WAVES_PER_BLK - 1) / WAVES_PER_BLK; // 391

    // graph normalization
    hipMemsetAsync(deg, 0, N_NODES * sizeof(float), stream);
    deg_kernel  <<<EB, TPB, 0, stream>>>(row, deg);
    dinv_kernel <<<NB, TPB, 0, stream>>>(deg, dinv);
    edgew_kernel<<<EB, TPB, 0, stream>>>(row, col, dinv, wE);

    // k=0: hid = x @ W0
    gemm_cheb<<<GB, WAVES_PER_BLK * 32, 0, stream>>>(x, cheb_w + 0 * D_IN * D_HID, hid, 0);

    // k=1: tx1 = prop(x) in bufA
    hipMemsetAsync(bufA, 0, (size_t)N_NODES * D_IN * sizeof(float), stream);
    scatter_kernel<<<SB, TPB, 0, stream>>>(row, col, wE, x, bufA, 1.0f);
    gemm_cheb<<<GB, WAVES_PER_BLK * 32, 0, stream>>>(bufA, cheb_w + 1 * D_IN * D_HID, hid, 1);

    // k=2: tx2 = 2*prop(tx1) - tx0  -> bufB
    negcopy_kernel<<<CB, TPB, 0, stream>>>((const float4*)x, (float4*)bufB);
    scatter_kernel<<<SB, TPB, 0, stream>>>(row, col, wE, bufA, bufB, 2.0f);
    gemm_cheb<<<GB, WAVES_PER_BLK * 32, 0, stream>>>(bufB, cheb_w + 2 * D_IN * D_HID, hid, 1);

    // k=3: tx3 = 2*prop(tx2) - tx1  -> bufC
    negcopy_kernel<<<CB, TPB, 0, stream>>>((const float4*)bufA, (float4*)bufC);
    scatter_kernel<<<SB, TPB, 0, stream>>>(row, col, wE, bufB, bufC, 2.0f);
    gemm_cheb<<<GB, WAVES_PER_BLK * 32, 0, stream>>>(bufC, cheb_w + 3 * D_IN * D_HID, hid, 1);

    // k=4: tx4 = 2*prop(tx3) - tx2  -> bufA (tx1 dead, reuse)
    negcopy_kernel<<<CB, TPB, 0, stream>>>((const float4*)bufB, (float4*)bufA);
    scatter_kernel<<<SB, TPB, 0, stream>>>(row, col, wE, bufC, bufA, 2.0f);
    gemm_cheb<<<GB, WAVES_PER_BLK * 32, 0, stream>>>(bufA, cheb_w + 4 * D_IN * D_HID, hid, 1);

    // head
    head_kernel<<<NB, TPB, 0, stream>>>(hid, cheb_b, fc_w, fc_b, out);
}